// VectorQuantizer_73280732004366
// MI455X (gfx1250) — compile-verified
//
#include <hip/hip_runtime.h>

typedef __attribute__((ext_vector_type(2))) float v2f;
typedef __attribute__((ext_vector_type(8))) float v8f;

#define NROWS   65536      // B*H*W
#define CDIM    64
#define KCODES  1024
#define KTILES  (KCODES / 16)
#define Q_ELEMS 4194304    // B*C*H*W
#define BSTRIDE 68         // padded LDS row stride (floats) -> conflict-free ds_load_b64

// ---------------------------------------------------------------- e2[k] = ||emb_k||^2
__global__ __launch_bounds__(256) void vq_e2_kernel(const float* __restrict__ emb,
                                                    float* __restrict__ e2) {
    int k = blockIdx.x * 256 + threadIdx.x;          // 4 blocks x 256 = 1024
    const float4* row = reinterpret_cast<const float4*>(emb + k * CDIM);
    float s = 0.f;
#pragma unroll
    for (int i = 0; i < 16; ++i) {
        float4 v = row[i];
        s += v.x * v.x + v.y * v.y + v.z * v.z + v.w * v.w;
    }
    e2[k] = s;
}

// ---------------------------------------------------------------- main: GEMM+argmin+gather
__global__ __launch_bounds__(256) void vq_main_kernel(const float* __restrict__ z,
                                                      const float* __restrict__ emb,
                                                      const float* __restrict__ e2g,
                                                      float* __restrict__ out,
                                                      float* __restrict__ partial) {
    __shared__ __align__(16) float b_sh[16 * BSTRIDE];   // one 16x64 codebook tile, padded
    __shared__ float e2_sh[KCODES];
    __shared__ int   idx_sh[8][16];
    __shared__ float warr[8];

    const int tid   = threadIdx.x;
    const int lane  = tid & 31;
    const int wv    = tid >> 5;            // wave in block (0..7)
    const int wlane = lane & 15;           // N / M position within half
    const int half  = lane >> 4;           // lane-half selects K pair

    // stage ||e||^2 into LDS once
#pragma unroll
    for (int i = 0; i < 4; ++i) e2_sh[tid + i * 256] = e2g[tid + i * 256];

    // this wave's 16 rows: n = b*4096 + h*64 + w,  w in [w0, w0+16)
    const int waveG = blockIdx.x * 8 + wv;
    const int n0    = waveG * 16;
    const int bb    = n0 >> 12;
    const int hh    = (n0 >> 6) & 63;
    const int w0    = n0 & 63;
    const int zbase = bb * 262144 + hh * 64 + w0 + wlane;   // + c*4096 selects channel

    // A fragments: lane holds row m=wlane, K = 4*c0 + 2*half + {0,1}  (WMMA f32 16x16x4 layout)
    v2f a[16];
#pragma unroll
    for (int c0 = 0; c0 < 16; ++c0) {
        int c = c0 * 4 + half * 2;
        a[c0].x = z[zbase + c * 4096];
        a[c0].y = z[zbase + (c + 1) * 4096];
    }

    // cooperative tile staging coords: 256 threads cover 16 rows x 64 cols (float4 each)
    const int srow = tid >> 4;
    const int scol = (tid & 15) * 4;

    float best[8];
    int   bidx[8];
#pragma unroll
    for (int r = 0; r < 8; ++r) { best[r] = 3.4e38f; bidx[r] = 0; }

    float4 stage = *reinterpret_cast<const float4*>(emb + srow * CDIM + scol); // tile 0

    for (int t = 0; t < KTILES; ++t) {
        __syncthreads();                                   // prev tile fully consumed
        *reinterpret_cast<float4*>(&b_sh[srow * BSTRIDE + scol]) = stage;
        if (t + 1 < KTILES)                                // prefetch next tile (hides latency)
            stage = *reinterpret_cast<const float4*>(emb + ((t + 1) * 16 + srow) * CDIM + scol);
        float e2v = e2_sh[t * 16 + wlane];
        __syncthreads();                                   // tile t visible

        v8f acc = {};
#pragma unroll
        for (int c0 = 0; c0 < 16; ++c0) {
            // B fragment: lane holds code n=wlane, K = 4*c0 + 2*half + {0,1} (conflict-free b64)
            v2f bf = *reinterpret_cast<const v2f*>(&b_sh[wlane * BSTRIDE + c0 * 4 + half * 2]);
            acc = __builtin_amdgcn_wmma_f32_16x16x4_f32(false, a[c0], false, bf,
                                                        (short)0, acc, false, false);
        }

        const int cand = t * 16 + wlane;
#pragma unroll
        for (int r = 0; r < 8; ++r) {
            float d = __builtin_fmaf(-2.0f, acc[r], e2v);  // ||e||^2 - 2 z.e  (argmin-equiv)
            if (d < best[r]) { best[r] = d; bidx[r] = cand; }
        }
    }

    // cross-lane argmin within each 16-lane group (first-min tie-break)
#pragma unroll
    for (int r = 0; r < 8; ++r) {
        float bv = best[r];
        int   bi = bidx[r];
#pragma unroll
        for (int m = 1; m < 16; m <<= 1) {
            float ob = __shfl_xor(bv, m, 32);
            int   oi = __shfl_xor(bi, m, 32);
            if (ob < bv || (ob == bv && oi < bi)) { bv = ob; bi = oi; }
        }
        if (wlane == 0) idx_sh[wv][r + 8 * half] = bi;     // lanes 0 and 16 publish rows r / r+8
    }
    __syncthreads();

    // idx output (as float, after q and the two scalar losses)
    float* outIdx = out + Q_ELEMS + 2;
    if (half == 0) outIdx[n0 + wlane] = (float)idx_sh[wv][wlane];

    // gather winning codes, write q (coalesced per channel), accumulate loss
    const int myIdx = idx_sh[wv][wlane];
    const float* erow = emb + myIdx * CDIM;
    float lsum = 0.f;
#pragma unroll 8
    for (int c2 = 0; c2 < 32; ++c2) {
        int c = c2 * 2 + half;
        float val = erow[c];
        int addr = zbase + c * 4096;
        float dv = val - z[addr];
        lsum = __builtin_fmaf(dv, dv, lsum);
        out[addr] = val;
    }

    // deterministic loss reduction: wave shfl -> LDS -> one partial per block
#pragma unroll
    for (int m = 16; m >= 1; m >>= 1) lsum += __shfl_xor(lsum, m, 32);
    if (lane == 0) warr[wv] = lsum;
    __syncthreads();
    if (tid == 0) {
        float s = 0.f;
#pragma unroll
        for (int i = 0; i < 8; ++i) s += warr[i];
        partial[blockIdx.x] = s;
    }
}

// ---------------------------------------------------------------- final loss reduction
__global__ __launch_bounds__(256) void vq_fin_kernel(const float* __restrict__ partial,
                                                     float* __restrict__ out) {
    __shared__ float sh[256];
    int t = threadIdx.x;
    sh[t] = partial[t] + partial[t + 256];                 // 512 partials
    __syncthreads();
    for (int s = 128; s > 0; s >>= 1) {
        if (t < s) sh[t] += sh[t + s];
        __syncthreads();
    }
    if (t == 0) {
        float loss = sh[0] * (1.0f / (float)Q_ELEMS);      // mean over B*H*W*C
        out[Q_ELEMS]     = loss;                           // codebook_loss
        out[Q_ELEMS + 1] = loss;                           // commitment_loss (same value)
    }
}

extern "C" void kernel_launch(void* const* d_in, const int* in_sizes, int n_in,
                              void* d_out, int out_size, void* d_ws, size_t ws_size,
                              hipStream_t stream) {
    const float* z   = (const float*)d_in[0];   // [16,64,64,64] f32
    const float* emb = (const float*)d_in[1];   // [1024,64]     f32
    float* out = (float*)d_out;                 // q | codebook | commit | idx(as f32)

    float* e2      = (float*)d_ws;              // 1024 floats
    float* partial = e2 + 1024;                 // 512 floats

    vq_e2_kernel  <<<4,   256, 0, stream>>>(emb, e2);
    vq_main_kernel<<<512, 256, 0, stream>>>(z, emb, e2, out, partial);
    vq_fin_kernel <<<1,   256, 0, stream>>>(partial, out);
}